// MULTIHEADSELFATTN_31361851195401
// MI455X (gfx1250) — compile-verified
//
#include <hip/hip_runtime.h>

// MI455X head-axis attention: B=4, S=4096, H=16, D=128.
// Memory-bound (~553 MB @ 23.3 TB/s ~ 24us). One wave32 per (b,s) tile:
//   S = Q*K^T/sqrt(2048) via 32x v_wmma_f32_16x16x4_f32
//   row-softmax in C-layout via 16-lane shfl_xor butterflies
//   +(-1e15) above diagonal (AFTER softmax, faithful to reference), store scores
//   O = P*V via 32x v_wmma_f32_16x16x4_f32 (P rearranged through LDS)
// Outputs written with non-temporal stores (never re-read).

typedef __attribute__((ext_vector_type(2))) float v2f;
typedef __attribute__((ext_vector_type(8))) float v8f;

#define WAVES_PER_BLOCK 8
#define NEG_BIG (-1e15f)

__global__ __launch_bounds__(32 * WAVES_PER_BLOCK)
void mhsa_head_attn_kernel(const float* __restrict__ Q,
                           const float* __restrict__ K,
                           const float* __restrict__ V,
                           float* __restrict__ Sout,   // [nTiles][16][16]
                           float* __restrict__ Oout,   // [nTiles][16][128]
                           int nTiles)
{
    // 16x16 P tile per wave, padded to stride 18 (even -> 8B-aligned float2
    // reads since col offsets are even; stride 18 avoids bank conflicts).
    __shared__ float plds[WAVES_PER_BLOCK * 16 * 18];

    const int wid  = threadIdx.x >> 5;
    const int lane = threadIdx.x & 31;
    const int row  = lane & 15;   // M (A) or N (B/C) index
    const int grp  = lane >> 4;   // half-wave group (K-pair / row-half select)
    const int tile = blockIdx.x * WAVES_PER_BLOCK + wid;
    if (tile >= nTiles) return;   // wave-uniform: EXEC stays all-ones for WMMA

    const float* Qt = Q + (size_t)tile * 2048;  // 16 rows x 128 cols
    const float* Kt = K + (size_t)tile * 2048;
    const float* Vt = V + (size_t)tile * 2048;

    // ---------- scores = Q(16x128) @ K^T(128x16), K accumulated 4 at a time.
    // A-frag lane L: Q[L&15][k0 + 2*(L>>4) + {0,1}]
    // B-frag (B = K^T) lane L: K[L&15][k0 + 2*(L>>4) + {0,1}]  (same pattern)
    v8f acc = {};
    const float* qrow = Qt + row * 128 + 2 * grp;
    const float* krow = Kt + row * 128 + 2 * grp;
#pragma unroll
    for (int k0 = 0; k0 < 128; k0 += 4) {
        v2f a = *(const v2f*)(qrow + k0);
        v2f b = *(const v2f*)(krow + k0);
        acc = __builtin_amdgcn_wmma_f32_16x16x4_f32(
                  false, a, false, b, (short)0, acc, false, false);
    }

    // ---------- scale + row softmax + post-softmax causal mask.
    // C/D layout: VGPR i, lanes 0-15 -> row h=i, lanes 16-31 -> row h=i+8,
    // col t = lane&15.  Softmax axis is over t -> reduce within 16-lane halves.
    const float inv_scale = 0.02209708691207961f;  // 1/sqrt(2048)
#pragma unroll
    for (int i = 0; i < 8; ++i) {
        float x = acc[i] * inv_scale;
        float m = x;
#pragma unroll
        for (int d = 1; d < 16; d <<= 1)
            m = fmaxf(m, __shfl_xor(m, d, 16));
        float e = __expf(x - m);
        float s = e;
#pragma unroll
        for (int d = 1; d < 16; d <<= 1)
            s += __shfl_xor(s, d, 16);
        float p = e / s;
        const int h = i + 8 * grp;
        if (row > h) p += NEG_BIG;   // mask added AFTER softmax (faithful)
        acc[i] = p;
    }

    // ---------- store masked scores (output #1), non-temporal.
    {
        float* sp = Sout + (size_t)tile * 256 + row;
#pragma unroll
        for (int i = 0; i < 8; ++i)
            __builtin_nontemporal_store(acc[i], sp + (size_t)(i + 8 * grp) * 16);
    }

    // ---------- rearrange P from C-layout to A-layout through LDS.
    float* pw = plds + wid * (16 * 18);
#pragma unroll
    for (int i = 0; i < 8; ++i)
        pw[(i + 8 * grp) * 18 + row] = acc[i];
    // same-wave LDS RAW: compiler inserts s_wait_dscnt; no barrier needed.

    v2f afrag[4];
#pragma unroll
    for (int k = 0; k < 4; ++k)
        afrag[k] = *(const v2f*)(pw + row * 18 + 4 * k + 2 * grp);

    // ---------- O = P(16x16) @ V(16x128): 8 column tiles x 4 K-steps.
    // B-frag lane L: V[k0+2*(L>>4)+{0,1}][nt*16 + (L&15)] (two rows, same col).
    float* obase = Oout + (size_t)tile * 2048 + row;
#pragma unroll
    for (int nt = 0; nt < 8; ++nt) {
        const float* vb = Vt + nt * 16 + row;
        v8f o = {};
#pragma unroll
        for (int k = 0; k < 4; ++k) {
            const int c = 4 * k + 2 * grp;
            v2f b;
            b.x = vb[c * 128];
            b.y = vb[(c + 1) * 128];
            o = __builtin_amdgcn_wmma_f32_16x16x4_f32(
                    false, afrag[k], false, b, (short)0, o, false, false);
        }
        float* op = obase + nt * 16;
#pragma unroll
        for (int i = 0; i < 8; ++i)
            __builtin_nontemporal_store(o[i], op + (size_t)(i + 8 * grp) * 128);
    }
}

extern "C" void kernel_launch(void* const* d_in, const int* in_sizes, int n_in,
                              void* d_out, int out_size, void* d_ws, size_t ws_size,
                              hipStream_t stream) {
    (void)n_in; (void)out_size; (void)d_ws; (void)ws_size;
    const float* Q = (const float*)d_in[0];
    const float* K = (const float*)d_in[1];
    const float* V = (const float*)d_in[2];

    const int nTiles = in_sizes[0] / 2048;           // B*S (H*D = 16*128 = 2048)
    float* Sout = (float*)d_out;                     // nTiles*256 floats
    float* Oout = Sout + (size_t)nTiles * 256;       // nTiles*2048 floats

    const int blocks = (nTiles + WAVES_PER_BLOCK - 1) / WAVES_PER_BLOCK;
    mhsa_head_attn_kernel<<<blocks, 32 * WAVES_PER_BLOCK, 0, stream>>>(
        Q, K, V, Sout, Oout, nTiles);
}